// GraphSAGEModel_52682068853204
// MI455X (gfx1250) — compile-verified
//
#include <hip/hip_runtime.h>
#include <hip/hip_bf16.h>

// ---------------------------------------------------------------------------
// GraphSAGE forward for MI455X (gfx1250).
// Strategy: atomic scatter-mean (L2-resident on 192MB L2), then f32 WMMA
// (V_WMMA_F32_16X16X4_F32) GEMMs, one wave32 per 16x16 output tile.
// ---------------------------------------------------------------------------

typedef float v2f __attribute__((ext_vector_type(2)));
typedef float v8f __attribute__((ext_vector_type(8)));

static constexpr int NN   = 100000;   // nodes
static constexpr int NE   = 1600000;  // edges
static constexpr int NG   = 64;       // graphs
static constexpr int DIN  = 128;
static constexpr int HID  = 256;
static constexpr int NATTR = 8;

// ---------------- utility ----------------
__global__ void zero_f32(float* __restrict__ p, long n) {
    long i = (long)blockIdx.x * blockDim.x + threadIdx.x;
    long stride = (long)gridDim.x * blockDim.x;
    for (; i < n; i += stride) p[i] = 0.0f;
}

// ---------------- degree count ----------------
__global__ void edge_deg(const long long* __restrict__ ei, float* __restrict__ deg) {
    int e = blockIdx.x * blockDim.x + threadIdx.x;
    if (e >= NE) return;
    int dst = (int)ei[NE + e];
    atomicAdd(&deg[dst], 1.0f);
}

// ---------------- scatter-add of source rows into destination rows --------
// 32 lanes per edge, D/32 contiguous floats per lane: coalesced row read,
// contiguous atomic burst into one destination row (L2-resident).
template <int D>
__global__ void scatter_add(const long long* __restrict__ ei,
                            const float* __restrict__ feat,
                            float* __restrict__ agg) {
    int t = blockIdx.x * blockDim.x + threadIdx.x;
    int e = t >> 5;
    int lane = t & 31;
    if (e >= NE) return;
    int src = (int)ei[e];
    int dst = (int)ei[NE + e];
    constexpr int PER = D / 32;
    const float* xs = feat + (size_t)src * D + lane * PER;
    float* ad = agg + (size_t)dst * D + lane * PER;
#pragma unroll
    for (int i = 0; i < PER; ++i) atomicAdd(&ad[i], xs[i]);
}

// ---------------- divide aggregated sums by degree ----------------
template <int SHIFT>  // D = 1<<SHIFT
__global__ void normalize_agg(float* __restrict__ agg, const float* __restrict__ deg) {
    long n = (long)NN << SHIFT;
    long i = (long)blockIdx.x * blockDim.x + threadIdx.x;
    long stride = (long)gridDim.x * blockDim.x;
    for (; i < n; i += stride) {
        float d = deg[i >> SHIFT];
        agg[i] = agg[i] / fmaxf(d, 1.0f);
    }
}

// ---------------- SAGE GEMM: C = act(A@Wl + b + X@Wr) ----------------
// One wave per 16x16 tile of C [NN x HID]. K-loop in steps of 4 using
// V_WMMA_F32_16X16X4_F32 (full f32 precision — workload is atomic-bound,
// so no reason to downcast).
// A 16x4 f32 layout: lane L holds row M=L%16; VGPR0 = K = k+2*(L/16),
// VGPR1 = K = k+2*(L/16)+1 (contiguous float2 from the A row).
// B 4x16: VGPR0 = row K=k+2*(L/16) at N=L%16, VGPR1 = next row.
// C/D 16x16: VGPR i -> row M = i + 8*(L/16), col N = L%16.
template <int K, bool RELU>
__global__ __launch_bounds__(256) void sage_gemm(
    const float* __restrict__ A, const float* __restrict__ X,
    const float* __restrict__ Wl, const float* __restrict__ Wr,
    const float* __restrict__ bias, float* __restrict__ C) {
    const int wave = threadIdx.x >> 5;
    const int lane = threadIdx.x & 31;
    const int gid  = blockIdx.x * 8 + wave;   // tile id: 0 .. NN/16*16 - 1
    const int tn   = gid & 15;                // 16 tiles across HID=256
    const int tm   = gid >> 4;                // 6250 tiles down NN=100000
    const int lo   = lane & 15;
    const int hi   = lane >> 4;
    const int n    = tn * 16 + lo;            // output column

    const float* arow = A + (size_t)(tm * 16 + lo) * K;
    const float* xrow = X + (size_t)(tm * 16 + lo) * K;

    v8f acc = {};
#pragma unroll 4
    for (int k = 0; k < K; k += 4) {
        v2f a, b;
        a.x = arow[k + 2 * hi];
        a.y = arow[k + 2 * hi + 1];
        b.x = Wl[(size_t)(k + 2 * hi) * HID + n];
        b.y = Wl[(size_t)(k + 2 * hi + 1) * HID + n];
        acc = __builtin_amdgcn_wmma_f32_16x16x4_f32(false, a, false, b,
                                                    (short)0, acc, false, false);
    }
#pragma unroll 4
    for (int k = 0; k < K; k += 4) {
        v2f a, b;
        a.x = xrow[k + 2 * hi];
        a.y = xrow[k + 2 * hi + 1];
        b.x = Wr[(size_t)(k + 2 * hi) * HID + n];
        b.y = Wr[(size_t)(k + 2 * hi + 1) * HID + n];
        acc = __builtin_amdgcn_wmma_f32_16x16x4_f32(false, a, false, b,
                                                    (short)0, acc, false, false);
    }

    const float bv = bias[n];
#pragma unroll
    for (int i = 0; i < 8; ++i) {
        float v = acc[i] + bv;
        if (RELU) v = fmaxf(v, 0.0f);
        C[(size_t)(tm * 16 + hi * 8 + i) * HID + n] = v;
    }
}

// ---------------- global mean pool ----------------
__global__ void pool_count(const long long* __restrict__ batch, float* __restrict__ gcnt) {
    int nidx = blockIdx.x * blockDim.x + threadIdx.x;
    if (nidx >= NN) return;
    atomicAdd(&gcnt[(int)batch[nidx]], 1.0f);
}

__global__ void pool_sum(const long long* __restrict__ batch,
                         const float* __restrict__ h2, float* __restrict__ gsum) {
    long total = (long)NN * HID;
    long i = (long)blockIdx.x * blockDim.x + threadIdx.x;
    long stride = (long)gridDim.x * blockDim.x;
    for (; i < total; i += stride) {
        int nidx = (int)(i >> 8);  // HID == 256
        int f = (int)(i & 255);
        atomicAdd(&gsum[(size_t)batch[nidx] * HID + f], h2[i]);
    }
}

// ---------------- head MLP ----------------
// gh = relu([gsum/gcnt, graph_attr] @ Wf1 + bf1);  64 blocks x 256 threads
__global__ void mlp1(const float* __restrict__ gsum, const float* __restrict__ gcnt,
                     const float* __restrict__ gattr, const float* __restrict__ Wf1,
                     const float* __restrict__ bf1, float* __restrict__ gh) {
    int g = blockIdx.x;    // graph
    int c = threadIdx.x;   // output column
    float inv = 1.0f / fmaxf(gcnt[g], 1.0f);
    float acc = bf1[c];
    for (int j = 0; j < HID; ++j)
        acc += (gsum[(size_t)g * HID + j] * inv) * Wf1[(size_t)j * HID + c];
    for (int j = 0; j < NATTR; ++j)
        acc += gattr[g * NATTR + j] * Wf1[(size_t)(HID + j) * HID + c];
    gh[(size_t)g * HID + c] = fmaxf(acc, 0.0f);
}

// out = gh @ Wf2 + bf2 ;  one block of 64 threads, dot over 256
__global__ void mlp2(const float* __restrict__ gh, const float* __restrict__ Wf2,
                     const float* __restrict__ bf2, float* __restrict__ out) {
    int g = threadIdx.x;
    if (g >= NG) return;
    float acc = bf2[0];
    for (int j = 0; j < HID; ++j) acc += gh[(size_t)g * HID + j] * Wf2[j];
    out[g] = acc;
}

// ---------------------------------------------------------------------------
extern "C" void kernel_launch(void* const* d_in, const int* in_sizes, int n_in,
                              void* d_out, int out_size, void* d_ws, size_t ws_size,
                              hipStream_t stream) {
    (void)in_sizes; (void)n_in; (void)out_size; (void)ws_size;

    const float*     x     = (const float*)d_in[0];
    const long long* ei    = (const long long*)d_in[1];   // int64 per reference
    const long long* batch = (const long long*)d_in[2];   // int64 per reference
    const float*     gattr = (const float*)d_in[3];
    const float*     W1l   = (const float*)d_in[4];
    const float*     b1    = (const float*)d_in[5];
    const float*     W1r   = (const float*)d_in[6];
    const float*     W2l   = (const float*)d_in[7];
    const float*     b2    = (const float*)d_in[8];
    const float*     W2r   = (const float*)d_in[9];
    const float*     Wf1   = (const float*)d_in[10];
    const float*     bf1   = (const float*)d_in[11];
    const float*     Wf2   = (const float*)d_in[12];
    const float*     bf2   = (const float*)d_in[13];
    float* out = (float*)d_out;

    // workspace layout (floats) — agg1, deg, agg2 contiguous so one zero pass
    float* ws   = (float*)d_ws;
    float* agg1 = ws;                                 // NN*DIN
    float* deg  = agg1 + (size_t)NN * DIN;            // NN
    float* agg2 = deg + NN;                           // NN*HID
    float* h1   = agg2 + (size_t)NN * HID;            // NN*HID
    float* h2   = h1 + (size_t)NN * HID;              // NN*HID
    float* gsum = h2 + (size_t)NN * HID;              // NG*HID
    float* gcnt = gsum + (size_t)NG * HID;            // NG
    float* gh   = gcnt + NG;                          // NG*HID

    // 1) zero all atomic targets (deterministic across replays)
    long zero1 = (long)NN * DIN + NN + (long)NN * HID;     // agg1+deg+agg2
    zero_f32<<<4096, 256, 0, stream>>>(agg1, zero1);
    zero_f32<<<64, 256, 0, stream>>>(gsum, (long)NG * HID + NG);

    // 2) degree counts
    edge_deg<<<(NE + 255) / 256, 256, 0, stream>>>(ei, deg);

    // 3) layer 1: scatter-mean of x, then WMMA GEMM + relu
    scatter_add<DIN><<<(NE * 32) / 256, 256, 0, stream>>>(ei, x, agg1);
    normalize_agg<7><<<8192, 256, 0, stream>>>(agg1, deg);
    sage_gemm<DIN, true><<<(NN / 16) * (HID / 16) / 8, 256, 0, stream>>>(
        agg1, x, W1l, W1r, b1, h1);

    // 4) layer 2: scatter-mean of h1, then WMMA GEMM (no relu)
    scatter_add<HID><<<(NE * 32) / 256, 256, 0, stream>>>(ei, h1, agg2);
    normalize_agg<8><<<8192, 256, 0, stream>>>(agg2, deg);
    sage_gemm<HID, false><<<(NN / 16) * (HID / 16) / 8, 256, 0, stream>>>(
        agg2, h1, W2l, W2r, b2, h2);

    // 5) global mean pool
    pool_count<<<(NN + 255) / 256, 256, 0, stream>>>(batch, gcnt);
    pool_sum<<<8192, 256, 0, stream>>>(batch, h2, gsum);

    // 6) head MLP
    mlp1<<<NG, HID, 0, stream>>>(gsum, gcnt, gattr, Wf1, bf1, gh);
    mlp2<<<1, 64, 0, stream>>>(gh, Wf2, bf2, out);
}